// MaskDecoder_37134287241628
// MI455X (gfx1250) — compile-verified
//
#include <hip/hip_runtime.h>

typedef float v2f __attribute__((ext_vector_type(2)));
typedef float v8f __attribute__((ext_vector_type(8)));

#define B_   4
#define NC_  1024
#define NP_  32768
#define C_   256
#define DIN_ 448
#define XD_  704          // C_ + DIN_
#define EPS_ 1e-5f

// padded LDS strides (floats): ≡4 or 16 (mod 64 banks) to kill A-frag bank conflicts,
// while keeping 8B alignment for ds_load_2addr_b64 and 16B for b128 async stores.
#define XP_  708          // x tile stride   (708 % 64 == 4)
#define HP_  260          // hidden stride   (260 % 64 == 4)
#define WP_  272          // W panel stride  (272 % 64 == 16)
#define KCH_ 32           // k-rows per staged W chunk
#define WBUF_ (KCH_ * WP_) // floats per W buffer (8704)
#define MT_  64           // rows per block in fused kernel (4 row-tiles x 16)

__device__ __forceinline__ float gelu_f(float x) {
  return 0.5f * x * (1.0f + erff(x * 0.7071067811865475f));
}

// CDNA5 async global->LDS copy (ASYNCcnt path), 16B per issue.
__device__ __forceinline__ void async_b128(unsigned lds_byte_off, const float* g) {
  asm volatile("global_load_async_to_lds_b128 %0, %1, off"
               :: "v"(lds_byte_off), "v"(g) : "memory");
}
__device__ __forceinline__ void wait_async0() {
  asm volatile("s_wait_asynccnt 0x0" ::: "memory");
}
__device__ __forceinline__ unsigned lds_off(const float* p) {
  return (unsigned)(unsigned long long)p;  // low 32 bits of generic ptr = LDS byte offset
}

// One wave: 16 x (NT*16) tile of As(16xK, LDS, stride lda) @ Ws(KxN, stride ldw), K fixed = KCH_.
// A-frag: a[v]=A[m16][ka+v], ka=k0+2*half ; B-frag: b[v]=W[ka+v][n] ; D: c[v]=D[v+8*half][m16].
template <int NT>
__device__ __forceinline__ void wave_gemm_chunk(const float* As, int lda,
                                                const float* Ws, int ldw,
                                                int n_base, v8f acc[NT]) {
  const int lane = threadIdx.x & 31, half = lane >> 4, m16 = lane & 15;
  for (int k0 = 0; k0 < KCH_; k0 += 4) {
    const int ka = k0 + 2 * half;
    v2f a;
    a.x = As[m16 * lda + ka + 0];
    a.y = As[m16 * lda + ka + 1];
#pragma unroll
    for (int j = 0; j < NT; ++j) {
      const int n = n_base + j * 16 + m16;
      v2f b;
      b.x = Ws[(ka + 0) * ldw + n];
      b.y = Ws[(ka + 1) * ldw + n];
      acc[j] = __builtin_amdgcn_wmma_f32_16x16x4_f32(
          false, a, false, b, (short)0, acc[j], false, false);
    }
  }
}

// Variant with W in global memory (encoder; W stays L2-resident, small kernel).
template <int NT>
__device__ __forceinline__ void wave_gemm_g(const float* As, int lda,
                                            const float* __restrict__ W,
                                            int Kdim, int n_base, v8f acc[NT]) {
  const int lane = threadIdx.x & 31, half = lane >> 4, m16 = lane & 15;
  for (int k0 = 0; k0 < Kdim; k0 += 4) {
    const int ka = k0 + 2 * half;
    v2f a;
    a.x = As[m16 * lda + ka + 0];
    a.y = As[m16 * lda + ka + 1];
#pragma unroll
    for (int j = 0; j < NT; ++j) {
      const int n = n_base + j * 16 + m16;
      v2f b;
      b.x = W[(size_t)(ka + 0) * C_ + n];
      b.y = W[(size_t)(ka + 1) * C_ + n];
      acc[j] = __builtin_amdgcn_wmma_f32_16x16x4_f32(
          false, a, false, b, (short)0, acc[j], false, false);
    }
  }
}

// ---------------- K1: encoder MLP3 + dense_prompt -> src (B*NC_, C_) ----------------
#define XP1_ 452   // 452 % 64 == 4
__global__ __launch_bounds__(128) void enc_kernel(
    const float* __restrict__ pc_emb, const float* __restrict__ dense_prompt,
    const float* __restrict__ w0, const float* __restrict__ b0,
    const float* __restrict__ w1, const float* __restrict__ b1,
    const float* __restrict__ w2, const float* __restrict__ b2,
    float* __restrict__ src_out) {
  __shared__ float xs[16 * XP1_];
  __shared__ float h1[16 * HP_];
  __shared__ float h2[16 * HP_];
  const int tid = threadIdx.x;
  const int r0  = blockIdx.x * 16;

  {
    const int row = tid >> 3, c0 = tid & 7;
    const float* p = pc_emb + (size_t)(r0 + row) * DIN_;
    for (int c = c0; c < DIN_; c += 8) xs[row * XP1_ + c] = p[c];
  }
  __syncthreads();

  const int wave = tid >> 5, lane = tid & 31, half = lane >> 4, m16 = lane & 15;
  const int nb = wave * 64;

  {
    v8f acc[4] = {};
    wave_gemm_g<4>(xs, XP1_, w0, DIN_, nb, acc);
#pragma unroll
    for (int j = 0; j < 4; ++j)
#pragma unroll
      for (int v = 0; v < 8; ++v) {
        const int rr = v + 8 * half, col = nb + j * 16 + m16;
        h1[rr * HP_ + col] = fmaxf(acc[j][v] + b0[col], 0.0f);
      }
  }
  __syncthreads();
  {
    v8f acc[4] = {};
    wave_gemm_g<4>(h1, HP_, w1, C_, nb, acc);
#pragma unroll
    for (int j = 0; j < 4; ++j)
#pragma unroll
      for (int v = 0; v < 8; ++v) {
        const int rr = v + 8 * half, col = nb + j * 16 + m16;
        h2[rr * HP_ + col] = fmaxf(acc[j][v] + b1[col], 0.0f);
      }
  }
  __syncthreads();
  {
    v8f acc[4] = {};
    wave_gemm_g<4>(h2, HP_, w2, C_, nb, acc);
#pragma unroll
    for (int j = 0; j < 4; ++j)
#pragma unroll
      for (int v = 0; v < 8; ++v) {
        const int rr = v + 8 * half, col = nb + j * 16 + m16;
        const size_t gi = (size_t)(r0 + rr) * C_ + col;
        src_out[gi] = acc[j][v] + b2[col] + dense_prompt[gi];
      }
  }
}

// ---------------- K2: hypernetwork heads (batch-independent) + iou head ----------------
__global__ __launch_bounds__(256) void tok_kernel(
    const float* __restrict__ iou_token, const float* __restrict__ mask_tokens,
    const float* __restrict__ hw0, const float* __restrict__ hb0,
    const float* __restrict__ hw1, const float* __restrict__ hb1,
    const float* __restrict__ hw2, const float* __restrict__ hb2,
    const float* __restrict__ iw0, const float* __restrict__ ib0,
    const float* __restrict__ iw1, const float* __restrict__ ib1,
    const float* __restrict__ iw2, const float* __restrict__ ib2,
    float* __restrict__ hyper3, float* __restrict__ iou_out) {
  __shared__ float t0[C_], t1[C_];
  const int c = threadIdx.x;

  for (int m = 1; m < 4; ++m) {
    float s = hb0[m * C_ + c];
    for (int k = 0; k < C_; ++k) s += mask_tokens[m * C_ + k] * hw0[((size_t)m * C_ + k) * C_ + c];
    t0[c] = fmaxf(s, 0.0f);
    __syncthreads();
    s = hb1[m * C_ + c];
    for (int k = 0; k < C_; ++k) s += t0[k] * hw1[((size_t)m * C_ + k) * C_ + c];
    t1[c] = fmaxf(s, 0.0f);
    __syncthreads();
    s = hb2[m * C_ + c];
    for (int k = 0; k < C_; ++k) s += t1[k] * hw2[((size_t)m * C_ + k) * C_ + c];
    hyper3[(m - 1) * C_ + c] = s;
    __syncthreads();
  }
  float s = ib0[c];
  for (int k = 0; k < C_; ++k) s += iou_token[k] * iw0[k * C_ + c];
  t0[c] = fmaxf(s, 0.0f);
  __syncthreads();
  s = ib1[c];
  for (int k = 0; k < C_; ++k) s += t0[k] * iw1[k * C_ + c];
  t1[c] = fmaxf(s, 0.0f);
  __syncthreads();
  if (c < 3) {
    float p = ib2[c + 1];
    for (int k = 0; k < C_; ++k) p += t1[k] * iw2[k * 4 + (c + 1)];
    for (int b = 0; b < B_; ++b) iou_out[b * 3 + c] = p;
  }
}

// ---------------- K3: fused gather -> GEMM(704) -> LN -> gelu -> GEMM(256) -> gelu -> masks ----
// 64 rows/block, 256 threads = 8 waves: wave = (row-tile rt in 0..3) x (col-half ch in {0,1}).
// W panels (32 k-rows x 256) double-buffered in LDS via async global->LDS loads.
__global__ __launch_bounds__(256) void fused_kernel(
    const float* __restrict__ src, const float* __restrict__ pf_feat,
    const float* __restrict__ iwgt, const int* __restrict__ iidx,
    const float* __restrict__ up_w0, const float* __restrict__ up_b0,
    const float* __restrict__ ln_g, const float* __restrict__ ln_b,
    const float* __restrict__ up_w1, const float* __restrict__ up_b1,
    const float* __restrict__ hyper3, float* __restrict__ masks_out) {
  extern __shared__ float smem[];
  float* xs     = smem;              // MT_*XP_ = 45312 floats (x tile; later reused as upscaled, stride HP_)
  float* wp     = xs + MT_ * XP_;    // 2*WBUF_ = 17408 floats (double-buffered W panel)
  float* hs     = wp + 2 * WBUF_;    // MT_*HP_ = 16640 floats
  float* red    = hs + MT_ * HP_;    // 256
  float* mu_s   = red + 256;         // 64
  float* rstd_s = mu_s + 64;         // 64

  const int tid = threadIdx.x;
  const int r0  = blockIdx.x * MT_;
  const unsigned xs_base = lds_off(xs);
  const unsigned wp_base = lds_off(wp);

  const int row = tid >> 2, cs = tid & 3;   // 4 threads per row, 64 rows
  const int r = r0 + row;
  const int bb = r >> 15;                   // / NP_
  const int pp = r & (NP_ - 1);
  const size_t bp = (size_t)bb * NP_ + pp;

  // W panel staging split: 256 threads, each stages 32 floats (8 x b128) of a 32x256 panel.
  const int kr = tid >> 3;            // 0..31 panel row
  const int ns = (tid & 7) * 32;      // 0..224 column segment

  // --- async: W0 chunk 0 -> wp[0] ---
  {
    const float* g = up_w0 + (size_t)kr * C_ + ns;
    const unsigned l = wp_base + (unsigned)(kr * WP_ + ns) * 4u;
#pragma unroll
    for (int j = 0; j < 8; ++j) async_b128(l + j * 16u, g + j * 4);
  }
  // --- async: pf_feat row segment -> xs cols [256,704) ---
  {
    const float* pf = pf_feat + bp * DIN_ + cs * 112;
    const unsigned l = xs_base + (unsigned)(row * XP_ + C_ + cs * 112) * 4u;
#pragma unroll
    for (int j = 0; j < 28; ++j) async_b128(l + j * 16u, pf + j * 4);
  }
  // --- interp (gather of 3 src rows) -> xs cols [0,256) ---
  {
    const int*   ip = iidx + bp * 3;
    const float* wq = iwgt + bp * 3;
    const float w0v = wq[0], w1v = wq[1], w2v = wq[2];
    const float* s0 = src + ((size_t)bb * NC_ + ip[0]) * C_;
    const float* s1 = src + ((size_t)bb * NC_ + ip[1]) * C_;
    const float* s2 = src + ((size_t)bb * NC_ + ip[2]) * C_;
    for (int c = cs * 64; c < cs * 64 + 64; c += 4) {
      float4 a = *(const float4*)(s0 + c);
      float4 u = *(const float4*)(s1 + c);
      float4 w = *(const float4*)(s2 + c);
      float4 o;
      o.x = w0v * a.x + w1v * u.x + w2v * w.x;
      o.y = w0v * a.y + w1v * u.y + w2v * w.y;
      o.z = w0v * a.z + w1v * u.z + w2v * w.z;
      o.w = w0v * a.w + w1v * u.w + w2v * w.w;
      *(float4*)(xs + row * XP_ + c) = o;
    }
  }
  wait_async0();
  __syncthreads();

  const int wave = tid >> 5, lane = tid & 31, half = lane >> 4, m16 = lane & 15;
  const int rt = wave & 3;           // row tile (16 rows each, 4 tiles)
  const int nb = (wave >> 2) * 128;  // column half

  {  // GEMM1: x(64x704) @ up_w0 + b0 -> hs ; double-buffered W panels
    v8f acc[8] = {};
    for (int kc = 0; kc < XD_ / KCH_; ++kc) {
      if (kc + 1 < XD_ / KCH_) {  // prefetch next panel
        const float* g = up_w0 + (size_t)((kc + 1) * KCH_ + kr) * C_ + ns;
        const unsigned l = wp_base + (unsigned)(((kc + 1) & 1) * WBUF_ + kr * WP_ + ns) * 4u;
#pragma unroll
        for (int j = 0; j < 8; ++j) async_b128(l + j * 16u, g + j * 4);
      }
      wave_gemm_chunk<8>(xs + rt * 16 * XP_ + kc * KCH_, XP_,
                         wp + (kc & 1) * WBUF_, WP_, nb, acc);
      wait_async0();
      __syncthreads();
    }
#pragma unroll
    for (int j = 0; j < 8; ++j)
#pragma unroll
      for (int v = 0; v < 8; ++v) {
        const int rr = rt * 16 + v + 8 * half, col = nb + j * 16 + m16;
        hs[rr * HP_ + col] = acc[j][v] + up_b0[col];
      }
  }
  __syncthreads();

  // prefetch W1 chunk 0 while doing LayerNorm
  {
    const float* g = up_w1 + (size_t)kr * C_ + ns;
    const unsigned l = wp_base + (unsigned)(kr * WP_ + ns) * 4u;
#pragma unroll
    for (int j = 0; j < 8; ++j) async_b128(l + j * 16u, g + j * 4);
  }

  {  // LayerNorm over C_ then exact gelu, in place on hs (row = tid>>2, seg = tid&3, 64 elems each)
    float s = 0.0f;
    for (int c = cs * 64; c < cs * 64 + 64; ++c) s += hs[row * HP_ + c];
    red[row * 4 + cs] = s;
    __syncthreads();
    if (tid < MT_) {
      float m = 0.0f;
      for (int j = 0; j < 4; ++j) m += red[tid * 4 + j];
      mu_s[tid] = m * (1.0f / C_);
    }
    __syncthreads();
    const float mu = mu_s[row];
    float v = 0.0f;
    for (int c = cs * 64; c < cs * 64 + 64; ++c) {
      const float d = hs[row * HP_ + c] - mu;
      v += d * d;
    }
    red[row * 4 + cs] = v;
    __syncthreads();
    if (tid < MT_) {
      float m = 0.0f;
      for (int j = 0; j < 4; ++j) m += red[tid * 4 + j];
      rstd_s[tid] = rsqrtf(m * (1.0f / C_) + EPS_);
    }
    __syncthreads();
    const float rs = rstd_s[row];
    for (int c = cs; c < C_; c += 4) {
      const float x = (hs[row * HP_ + c] - mu) * rs * ln_g[c] + ln_b[c];
      hs[row * HP_ + c] = gelu_f(x);
    }
  }
  wait_async0();
  __syncthreads();

  float* us = xs;  // reuse x-tile LDS for upscaled (MT_ x C_, stride HP_)
  {  // GEMM2: hs(64x256) @ up_w1 + b1, gelu -> us ; double-buffered W panels
    v8f acc[8] = {};
    for (int kc = 0; kc < C_ / KCH_; ++kc) {
      if (kc + 1 < C_ / KCH_) {
        const float* g = up_w1 + (size_t)((kc + 1) * KCH_ + kr) * C_ + ns;
        const unsigned l = wp_base + (unsigned)(((kc + 1) & 1) * WBUF_ + kr * WP_ + ns) * 4u;
#pragma unroll
        for (int j = 0; j < 8; ++j) async_b128(l + j * 16u, g + j * 4);
      }
      wave_gemm_chunk<8>(hs + rt * 16 * HP_ + kc * KCH_, HP_,
                         wp + (kc & 1) * WBUF_, WP_, nb, acc);
      wait_async0();
      __syncthreads();
    }
#pragma unroll
    for (int j = 0; j < 8; ++j)
#pragma unroll
      for (int v = 0; v < 8; ++v) {
        const int rr = rt * 16 + v + 8 * half, col = nb + j * 16 + m16;
        us[rr * HP_ + col] = gelu_f(acc[j][v] + up_b1[col]);
      }
  }
  __syncthreads();

  if (tid < 3 * MT_) {  // masks[b][m][p] = hyper3[m] . upscaled[row]
    const int mrow = tid & (MT_ - 1), m = tid >> 6;
    const int rr = r0 + mrow;
    const int b = rr >> 15;
    const int p = rr & (NP_ - 1);
    float s = 0.0f;
    for (int c = 0; c < C_; ++c) s += hyper3[m * C_ + c] * us[mrow * HP_ + c];
    masks_out[((size_t)(b * 3 + m)) * NP_ + p] = s;
  }
}

extern "C" void kernel_launch(void* const* d_in, const int* in_sizes, int n_in,
                              void* d_out, int out_size, void* d_ws, size_t ws_size,
                              hipStream_t stream) {
  const float* pc_embeddings = (const float*)d_in[0];
  /* d_in[1] pc_pe and d_in[2] sparse_prompt are dead inputs */
  const float* dense_prompt  = (const float*)d_in[3];
  const float* pf_feat       = (const float*)d_in[4];
  const float* interp_weight = (const float*)d_in[5];
  const int*   interp_index  = (const int*)d_in[6];
  const float* iou_token     = (const float*)d_in[7];
  const float* mask_tokens   = (const float*)d_in[8];
  const float* enc_w0 = (const float*)d_in[9];
  const float* enc_b0 = (const float*)d_in[10];
  const float* enc_w1 = (const float*)d_in[11];
  const float* enc_b1 = (const float*)d_in[12];
  const float* enc_w2 = (const float*)d_in[13];
  const float* enc_b2 = (const float*)d_in[14];
  const float* up_w0  = (const float*)d_in[15];
  const float* up_b0  = (const float*)d_in[16];
  const float* ln_g   = (const float*)d_in[17];
  const float* ln_b   = (const float*)d_in[18];
  const float* up_w1  = (const float*)d_in[19];
  const float* up_b1  = (const float*)d_in[20];
  const float* hyp_w0 = (const float*)d_in[21];
  const float* hyp_b0 = (const float*)d_in[22];
  const float* hyp_w1 = (const float*)d_in[23];
  const float* hyp_b1 = (const float*)d_in[24];
  const float* hyp_w2 = (const float*)d_in[25];
  const float* hyp_b2 = (const float*)d_in[26];
  const float* iou_w0 = (const float*)d_in[27];
  const float* iou_b0 = (const float*)d_in[28];
  const float* iou_w1 = (const float*)d_in[29];
  const float* iou_b1 = (const float*)d_in[30];
  const float* iou_w2 = (const float*)d_in[31];
  const float* iou_b2 = (const float*)d_in[32];

  float* ws     = (float*)d_ws;
  float* src    = ws;                         // B_*NC_*C_ floats (4 MB)
  float* hyper3 = ws + (size_t)B_ * NC_ * C_; // 3*C_ floats
  float* masks  = (float*)d_out;              // B_*3*NP_
  float* iou_o  = masks + (size_t)B_ * 3 * NP_;

  enc_kernel<<<(B_ * NC_) / 16, 128, 0, stream>>>(
      pc_embeddings, dense_prompt, enc_w0, enc_b0, enc_w1, enc_b1, enc_w2, enc_b2, src);

  tok_kernel<<<1, 256, 0, stream>>>(
      iou_token, mask_tokens, hyp_w0, hyp_b0, hyp_w1, hyp_b1, hyp_w2, hyp_b2,
      iou_w0, iou_b0, iou_w1, iou_b1, iou_w2, iou_b2, hyper3, iou_o);

  const size_t smem_bytes =
      (size_t)(MT_ * XP_ + 2 * WBUF_ + MT_ * HP_ + 256 + 64 + 64) * sizeof(float);
  fused_kernel<<<(B_ * NP_) / MT_, 256, smem_bytes, stream>>>(
      src, pf_feat, interp_weight, interp_index,
      up_w0, up_b0, ln_g, ln_b, up_w1, up_b1, hyper3, masks);
}